// RGCN_3298534884295
// MI455X (gfx1250) — compile-verified
//
#include <hip/hip_runtime.h>

typedef float v8f __attribute__((ext_vector_type(8)));
typedef float v2f __attribute__((ext_vector_type(2)));
typedef unsigned int v4u __attribute__((ext_vector_type(4)));
typedef int v8i __attribute__((ext_vector_type(8)));
typedef int v4i __attribute__((ext_vector_type(4)));

#define BM 64          // rows per block tile
#define KC 64          // K chunk staged in LDS
#define RS 68          // padded LDS row stride (floats): conflict-free, 16B rows

#if defined(__has_builtin)
#if __has_builtin(__builtin_amdgcn_tensor_load_to_lds) && __has_builtin(__builtin_amdgcn_s_wait_tensorcnt)
#define HAVE_TDM 1
#endif
#endif
#ifndef HAVE_TDM
#define HAVE_TDM 0
#endif

#if HAVE_TDM
// Issue one TDM load of a BM x KC fp32 tile starting at gsrc (row stride = lda
// elements) into LDS at lds_off_bytes. TDM pad feature inserts 4 DWORDs after
// every 64 DWORDs stored -> effective LDS row stride = RS floats (bank-safe).
// rows_rem / cols_rem give tile-relative tensor dims so the M-tail reads as 0.
__device__ __forceinline__ void tdm_load_tile(
    const float* gsrc, unsigned lds_off_bytes, int rows_rem, int cols_rem, int lda)
{
    // ---- D# group 0 ----
    v4u g0;
    g0[0] = 1u;                                   // count=1, user descriptor
    g0[1] = lds_off_bytes;                        // lds_addr (bytes)
    unsigned long long ga = (unsigned long long)(uintptr_t)gsrc;
    g0[2] = (unsigned)(ga & 0xFFFFFFFFull);       // global_addr[31:0]
    g0[3] = (unsigned)((ga >> 32) & 0x1FFFFFFull) // global_addr[56:32]
          | 0x80000000u;                          // type = 2 ("image")

    // ---- D# group 1 ----
    v8i g1;
    g1[0] = (int)((2u << 16)     // data_size = 4 bytes
                | (1u << 20)     // pad_enable
                | (5u << 22)     // pad_interval: 64 DWORDs
                | (3u << 25));   // pad_amount:   4 DWORDs
    const unsigned td0 = (unsigned)cols_rem;      // tensor_dim0 (x OOB bound)
    const unsigned td1 = (unsigned)rows_rem;      // tensor_dim1 (y OOB bound)
    g1[1] = (int)((td0 & 0xFFFFu) << 16);
    g1[2] = (int)((td0 >> 16) | ((td1 & 0xFFFFu) << 16));
    g1[3] = (int)(((td1 >> 16) & 0xFFFFu) | ((unsigned)KC << 16)); // tile_dim0
    g1[4] = BM;                                   // tile_dim1 (tile_dim2 = 0)
    g1[5] = lda;                                  // tensor_dim0_stride (elems)
    g1[6] = 0;
    g1[7] = 0;

    const v4i z4 = {0, 0, 0, 0};
    const v8i z8 = {0, 0, 0, 0, 0, 0, 0, 0};
    __builtin_amdgcn_tensor_load_to_lds(g0, g1, z4, z4, z8, 0);
}
#endif

// ---------------------------------------------------------------------------
// fp32 WMMA GEMM:  C[M, 128*TPW] = act(A[M,K] @ B[K,NOUT] + bias)
// Block = 256 threads (8 waves). Block tile = 64 x NOUT; wave w owns N-tiles
// {w, w+8, w+16} x 4 M-tiles. A chunks double-buffered in LDS via TDM
// (fallback: manual float4 staging). B is pre-packed pair-interleaved
// Bp[k/2][n][2] so each WMMA B-fragment is one 8-byte global load.
// ---------------------------------------------------------------------------
template <int TPW>
__global__ __launch_bounds__(256) void gemm_wmma(
    const float* __restrict__ A, int lda,
    const float* __restrict__ Bp,         // packed: [K/2][NOUT][2]
    const float* __restrict__ bias,       // [NOUT]
    float* __restrict__ C, int ldc,
    int M, int K, int do_lrelu)
{
    constexpr int NOUT = TPW * 128;
    __shared__ float sA[2][BM * RS];      // 2 x 17 KB

    const int tid   = threadIdx.x;
    const int wave  = tid >> 5;
    const int lane  = tid & 31;
    const int m0    = blockIdx.x * BM;
    const int mrow  = lane & 15;
    const int khalf = (lane >> 4) << 1;   // 0 | 2

    v8f acc[TPW][4] = {};

#if HAVE_TDM
    if (tid < 32) {
        tdm_load_tile(A + (size_t)m0 * lda, (unsigned)(uintptr_t)&sA[0][0],
                      M - m0, K, lda);
        __builtin_amdgcn_s_wait_tensorcnt(0);
    }
    __syncthreads();
#endif

    int buf = 0;
    for (int k0 = 0; k0 < K; k0 += KC) {
#if HAVE_TDM
        // prefetch next chunk into the other buffer while we compute
        if ((k0 + KC < K) && tid < 32) {
            tdm_load_tile(A + (size_t)m0 * lda + (k0 + KC),
                          (unsigned)(uintptr_t)&sA[buf ^ 1][0],
                          M - m0, K - (k0 + KC), lda);
        }
#else
        #pragma unroll
        for (int idx = tid * 4; idx < BM * KC; idx += 256 * 4) {
            const int r = idx >> 6;       // / KC
            const int c = idx & (KC - 1);
            const int gm = m0 + r;
            float4 v = make_float4(0.f, 0.f, 0.f, 0.f);
            if (gm < M) v = *(const float4*)(A + (size_t)gm * lda + k0 + c);
            *(float4*)(&sA[buf][r * RS + c]) = v;
        }
        __syncthreads();
#endif

        for (int kk = 0; kk < KC; kk += 4) {
            const int kb = kk + khalf;

            v2f afrag[4];
            #pragma unroll
            for (int mt = 0; mt < 4; ++mt) {
                const float* ap = &sA[buf][(mt * 16 + mrow) * RS + kb];
                v2f a = { ap[0], ap[1] };
                afrag[mt] = a;
            }

            #pragma unroll
            for (int nt = 0; nt < TPW; ++nt) {
                const int ncol = ((nt * 8 + wave) << 4) + mrow;
                const v2f bfrag =
                    *(const v2f*)(Bp + ((size_t)((k0 + kb) >> 1) * NOUT + ncol) * 2);
                #pragma unroll
                for (int mt = 0; mt < 4; ++mt) {
                    acc[nt][mt] = __builtin_amdgcn_wmma_f32_16x16x4_f32(
                        false, afrag[mt], false, bfrag,
                        (short)0, acc[nt][mt], false, false);
                }
            }
        }

#if HAVE_TDM
        if (tid < 32) __builtin_amdgcn_s_wait_tensorcnt(0);
        __syncthreads();
        buf ^= 1;
#else
        __syncthreads();
#endif
    }

    // ---- epilogue: bias (+ leaky relu), store ----
    #pragma unroll
    for (int nt = 0; nt < TPW; ++nt) {
        const int ncol = ((nt * 8 + wave) << 4) + mrow;
        const float bv = bias[ncol];
        #pragma unroll
        for (int mt = 0; mt < 4; ++mt) {
            const int rbase = m0 + mt * 16 + ((lane >> 4) << 3);
            #pragma unroll
            for (int r = 0; r < 8; ++r) {
                const int row = rbase + r;
                if (row < M) {
                    float v = acc[nt][mt][r] + bv;
                    if (do_lrelu) v = (v >= 0.f) ? v : 0.01f * v;
                    C[(size_t)row * ldc + ncol] = v;
                }
            }
        }
    }
}

// ---------------------------------------------------------------------------
// Small helper kernels
// ---------------------------------------------------------------------------

// Build fused conv weights directly in pair-interleaved layout:
// w_catp[k/2][c][2] over logical [128,384] = [w_root | w_rel0 | w_rel1]
__global__ void pack_weights(const float* __restrict__ w_root,
                             const float* __restrict__ w_rel,
                             const float* __restrict__ b_conv,
                             float* __restrict__ w_catp,
                             float* __restrict__ b_cat)
{
    const int i = blockIdx.x * blockDim.x + threadIdx.x;
    if (i < 128 * 384) {
        const int k = i / 384, c = i % 384;
        float v;
        if (c < 128) v = w_root[k * 128 + c];
        else         v = w_rel[(c >= 256 ? 1 : 0) * 128 * 128 + k * 128 + (c & 127)];
        w_catp[((size_t)(k >> 1) * 384 + c) * 2 + (k & 1)] = v;
    }
    if (i < 384) b_cat[i] = (i < 128) ? b_conv[i] : 0.0f;
}

// Generic pair-interleave pack: Bp[k/2][n][2] = B[k][n]
__global__ void pack_pairs(const float* __restrict__ B, float* __restrict__ Bp,
                           int K, int N)
{
    const int i = blockIdx.x * blockDim.x + threadIdx.x;
    if (i < K * N) {
        const int k = i / N, n = i % N;
        Bp[((size_t)(k >> 1) * N + n) * 2 + (k & 1)] = B[i];
    }
}

__global__ void zero_f(float* __restrict__ p, int n) {
    const int i = blockIdx.x * blockDim.x + threadIdx.x;
    if (i < n) p[i] = 0.0f;
}

__global__ void edge_count(const int* __restrict__ edge_index,
                           const int* __restrict__ edge_type,
                           float* __restrict__ cnt, int E, int N)
{
    const int e = blockIdx.x * blockDim.x + threadIdx.x;
    if (e < E) {
        const int r   = edge_type[e];
        const int dst = edge_index[E + e];
        atomicAdd(&cnt[r * N + dst], 1.0f);
    }
}

__global__ void recip_max1(float* __restrict__ p, int n) {
    const int i = blockIdx.x * blockDim.x + threadIdx.x;
    if (i < n) p[i] = 1.0f / fmaxf(p[i], 1.0f);
}

// out[i,0:128] = y[i,0:128]   (root term of fused conv GEMM)
__global__ void copy_root(const float* __restrict__ y, float* __restrict__ out, int N) {
    const int i = blockIdx.x * blockDim.x + threadIdx.x;
    if (i < N * 128) {
        const int row = i >> 7, h = i & 127;
        out[i] = y[(size_t)row * 384 + h];
    }
}

// One wave per edge: out[dst] += inv_cnt[r,dst] * y[src, 128 + r*128 : +128]
__global__ __launch_bounds__(256) void rgcn_scatter(
    const float* __restrict__ y,          // [N,384]
    const int* __restrict__ edge_index,   // [2,E]
    const int* __restrict__ edge_type,    // [E]
    const float* __restrict__ inv_cnt,    // [2,N]
    float* __restrict__ out,              // [N,128]
    int E, int N)
{
    const int wv   = (blockIdx.x * blockDim.x + threadIdx.x) >> 5;
    const int lane = threadIdx.x & 31;
    if (wv >= E) return;
    const int src = edge_index[wv];
    const int dst = edge_index[E + wv];
    const int r   = edge_type[wv];
    const float s = inv_cnt[r * N + dst];

    const float4 v = *(const float4*)(y + (size_t)src * 384 + 128 + r * 128 + lane * 4);
    float* op = out + (size_t)dst * 128 + lane * 4;
    atomicAdd(op + 0, v.x * s);
    atomicAdd(op + 1, v.y * s);
    atomicAdd(op + 2, v.z * s);
    atomicAdd(op + 3, v.w * s);
}

// One wave per row: out[i,0:3] = x[i,:] @ w_out[128,3] + b_out
__global__ __launch_bounds__(256) void out_gemv(
    const float* __restrict__ x, const float* __restrict__ w,
    const float* __restrict__ b, float* __restrict__ out, int N)
{
    const int row  = (blockIdx.x * blockDim.x + threadIdx.x) >> 5;
    const int lane = threadIdx.x & 31;
    if (row >= N) return;
    const float4 v = *(const float4*)(x + (size_t)row * 128 + lane * 4);
    const int h = lane * 4;
    float s0 = v.x * w[h * 3 + 0] + v.y * w[(h + 1) * 3 + 0] + v.z * w[(h + 2) * 3 + 0] + v.w * w[(h + 3) * 3 + 0];
    float s1 = v.x * w[h * 3 + 1] + v.y * w[(h + 1) * 3 + 1] + v.z * w[(h + 2) * 3 + 1] + v.w * w[(h + 3) * 3 + 1];
    float s2 = v.x * w[h * 3 + 2] + v.y * w[(h + 1) * 3 + 2] + v.z * w[(h + 2) * 3 + 2] + v.w * w[(h + 3) * 3 + 2];
    #pragma unroll
    for (int off = 16; off > 0; off >>= 1) {
        s0 += __shfl_down(s0, off, 32);
        s1 += __shfl_down(s1, off, 32);
        s2 += __shfl_down(s2, off, 32);
    }
    if (lane == 0) {
        out[(size_t)row * 3 + 0] = s0 + b[0];
        out[(size_t)row * 3 + 1] = s1 + b[1];
        out[(size_t)row * 3 + 2] = s2 + b[2];
    }
}

static inline int cdiv(int a, int b) { return (a + b - 1) / b; }

extern "C" void kernel_launch(void* const* d_in, const int* in_sizes, int n_in,
                              void* d_out, int out_size, void* d_ws, size_t ws_size,
                              hipStream_t stream)
{
    const float* feature    = (const float*)d_in[0];
    const int*   edge_index = (const int*)  d_in[1];
    const int*   edge_type  = (const int*)  d_in[2];
    const float* w_in       = (const float*)d_in[3];
    const float* b_in       = (const float*)d_in[4];
    const float* w_rel      = (const float*)d_in[5];
    const float* w_root     = (const float*)d_in[6];
    const float* b_conv     = (const float*)d_in[7];
    const float* w_out      = (const float*)d_in[8];
    const float* b_out      = (const float*)d_in[9];
    float* out = (float*)d_out;

    const int N = in_sizes[0] / 768;      // 50000
    const int E = in_sizes[2];            // 600000

    // workspace layout (floats)
    float* ws     = (float*)d_ws;
    float* x      = ws;                               // [N,128]
    float* y      = x      + (size_t)N * 128;         // [N,384]
    float* cnt    = y      + (size_t)N * 384;         // [2,N]
    float* w_catp = cnt    + (size_t)2 * N;           // packed [128,384]
    float* w_inp  = w_catp + (size_t)128 * 384;       // packed [768,128]
    float* b_cat  = w_inp  + (size_t)768 * 128;       // [384]

    // graph-constant prep (shared by both conv layers)
    pack_weights<<<cdiv(128 * 384, 256), 256, 0, stream>>>(w_root, w_rel, b_conv, w_catp, b_cat);
    pack_pairs  <<<cdiv(768 * 128, 256), 256, 0, stream>>>(w_in, w_inp, 768, 128);
    zero_f<<<cdiv(2 * N, 256), 256, 0, stream>>>(cnt, 2 * N);
    edge_count<<<cdiv(E, 256), 256, 0, stream>>>(edge_index, edge_type, cnt, E, N);
    recip_max1<<<cdiv(2 * N, 256), 256, 0, stream>>>(cnt, 2 * N);

    const dim3 gblk(cdiv(N, BM));

    // x = leaky_relu(feature @ w_in + b_in)
    gemm_wmma<1><<<gblk, 256, 0, stream>>>(feature, 768, w_inp, b_in, x, 128, N, 768, 1);

    // two RGCN conv layers (shared weights)
    for (int it = 0; it < 2; ++it) {
        gemm_wmma<3><<<gblk, 256, 0, stream>>>(x, 128, w_catp, b_cat, y, 384, N, 128, 0);
        copy_root<<<cdiv(N * 128, 256), 256, 0, stream>>>(y, x, N);
        rgcn_scatter<<<cdiv(E, 8), 256, 0, stream>>>(y, edge_index, edge_type, cnt, x, E, N);
    }

    // out = x @ w_out + b_out
    out_gemv<<<cdiv(N, 8), 256, 0, stream>>>(x, w_out, b_out, out, N);
}